// E2N3DLayer_2190433321017
// MI455X (gfx1250) — compile-verified
//
#include <hip/hip_runtime.h>

#define B_DIM 8
#define M_IN 32
#define M_OUT 32
#define C_DIM 64
#define T_DIM 192   // 3 * C

typedef float v2f __attribute__((ext_vector_type(2)));
typedef float v8f __attribute__((ext_vector_type(8)));

// ---------------------------------------------------------------------------
// Kernel 1: per-(b,m) slab marginal reduction. 256 threads, 1 MB slab each.
// Thread layout: q = tid&15 -> k-quad (float4 over k), r = tid>>4 -> row set.
// Per (i,jj) iteration the block reads one fully contiguous 4 KB stripe.
// ---------------------------------------------------------------------------
__global__ __launch_bounds__(256) void e2n_reduce_kernel(
    const float* __restrict__ x, float* __restrict__ s)
{
    __shared__ float lds1[C_DIM];   // s1[i] : sum over j,k
    __shared__ float lds2[C_DIM];   // s2[j] : sum over i,k
    __shared__ float lds3[C_DIM];   // s3[k] : sum over i,j

    const int tid = threadIdx.x;
    if (tid < C_DIM) { lds1[tid] = 0.f; lds2[tid] = 0.f; lds3[tid] = 0.f; }
    __syncthreads();

    const int slab = blockIdx.x;                       // b*M_IN + m
    const float* xs = x + (size_t)slab * (C_DIM * C_DIM * C_DIM);

    const int q    = tid & 15;                         // k = 4q .. 4q+3
    const int r    = tid >> 4;                         // 0..15 (j base)
    const int lane = tid & 31;                         // wave32 lane

    float4 acc3 = make_float4(0.f, 0.f, 0.f, 0.f);     // per-k accumulators
    float  acc2[4] = {0.f, 0.f, 0.f, 0.f};             // per-j (j = r + 16*jj)

    for (int i = 0; i < C_DIM; ++i) {
        const float* xi = xs + (i << 12);
        float acc1 = 0.f;
        #pragma unroll
        for (int jj = 0; jj < 4; ++jj) {
            const int j = r + (jj << 4);
            float4 v = *(const float4*)(xi + (j << 6) + (q << 2));
            float  h = (v.x + v.y) + (v.z + v.w);
            acc1      += h;
            acc2[jj]  += h;
            acc3.x += v.x; acc3.y += v.y; acc3.z += v.z; acc3.w += v.w;
        }
        // full-wave reduction of acc1 (covers all k, 8 j-values per wave)
        #pragma unroll
        for (int off = 16; off >= 1; off >>= 1)
            acc1 += __shfl_xor(acc1, off, 32);
        if (lane == 0) atomicAdd(&lds1[i], acc1);      // 8 atomics per i
    }

    // s2: reduce over k within each 16-lane half (xor masks stay in-half)
    #pragma unroll
    for (int jj = 0; jj < 4; ++jj) {
        float v = acc2[jj];
        #pragma unroll
        for (int off = 8; off >= 1; off >>= 1)
            v += __shfl_xor(v, off, 32);
        if ((lane & 15) == 0) atomicAdd(&lds2[r + (jj << 4)], v);
    }

    // s3: fold upper half-wave (same q) onto lower, then per-k atomics
    acc3.x += __shfl_xor(acc3.x, 16, 32);
    acc3.y += __shfl_xor(acc3.y, 16, 32);
    acc3.z += __shfl_xor(acc3.z, 16, 32);
    acc3.w += __shfl_xor(acc3.w, 16, 32);
    if (lane < 16) {                                   // here q == lane
        atomicAdd(&lds3[(q << 2) + 0], acc3.x);
        atomicAdd(&lds3[(q << 2) + 1], acc3.y);
        atomicAdd(&lds3[(q << 2) + 2], acc3.z);
        atomicAdd(&lds3[(q << 2) + 3], acc3.w);
    }
    __syncthreads();

    if (tid < T_DIM) {
        float v = (tid < 64) ? lds1[tid]
                : (tid < 128) ? lds2[tid - 64]
                              : lds3[tid - 128];
        s[(size_t)slab * T_DIM + tid] = v;
    }
}

// ---------------------------------------------------------------------------
// Kernel 2: out[b,n,t] = sum_m wsum[m,n] * s[b,m,t]  via V_WMMA_F32_16X16X4_F32
// One wave (32 threads, EXEC all ones) per 16x16 output tile; 8 K-steps (K=32).
// A (16x4, 32-bit): lanes 0-15 -> K = k0+{0,1} in vgpr{0,1}; lanes 16-31 -> K+2.
// B (4x16): row K striped across lane half, same K split as A.
// C/D (16x16 f32): vgpr rr -> M = rr + 8*(lane>>4), N = lane&15.
// ---------------------------------------------------------------------------
__global__ __launch_bounds__(32) void e2n_wmma_mix_kernel(
    const float* __restrict__ s,
    const float* __restrict__ alpha,
    const float* __restrict__ beta,
    const float* __restrict__ gamma,
    float* __restrict__ out)
{
    const int lane = threadIdx.x;
    const int half = lane >> 4;        // 0 or 1
    const int l16  = lane & 15;

    const int bid    = blockIdx.x;     // B * 2 * 12 blocks
    const int b      = bid / 24;
    const int rem    = bid % 24;
    const int n_base = (rem / 12) << 4;    // output-row tile (n)
    const int t_base = (rem % 12) << 4;    // output-col tile (t = d*64+c)

    const float* sb = s + (size_t)b * (M_IN * T_DIM);

    v8f acc = {0.f, 0.f, 0.f, 0.f, 0.f, 0.f, 0.f, 0.f};

    #pragma unroll
    for (int k0 = 0; k0 < M_IN; k0 += 4) {
        const int m0 = k0 + (half << 1);   // this lane-half's K pair
        const int n  = n_base + l16;
        const int t  = t_base + l16;

        v2f a, bf;
        a.x = alpha[m0 * M_OUT + n] + beta[m0 * M_OUT + n] + gamma[m0 * M_OUT + n];
        a.y = alpha[(m0 + 1) * M_OUT + n] + beta[(m0 + 1) * M_OUT + n]
            + gamma[(m0 + 1) * M_OUT + n];
        bf.x = sb[m0 * T_DIM + t];
        bf.y = sb[(m0 + 1) * T_DIM + t];

        acc = __builtin_amdgcn_wmma_f32_16x16x4_f32(
            /*neg_a=*/false, a, /*neg_b=*/false, bf,
            /*c_mod=*/(short)0, acc, /*reuse_a=*/false, /*reuse_b=*/false);
    }

    #pragma unroll
    for (int rr = 0; rr < 8; ++rr) {
        const int n = n_base + rr + (half << 3);
        const int t = t_base + l16;
        out[((size_t)b * M_OUT + n) * T_DIM + t] = acc[rr];
    }
}

extern "C" void kernel_launch(void* const* d_in, const int* in_sizes, int n_in,
                              void* d_out, int out_size, void* d_ws, size_t ws_size,
                              hipStream_t stream) {
    const float* x     = (const float*)d_in[0];
    const float* alpha = (const float*)d_in[1];
    const float* beta  = (const float*)d_in[2];
    const float* gamma = (const float*)d_in[3];
    float* out = (float*)d_out;
    float* s   = (float*)d_ws;   // B*M_IN*192 floats = 192 KB scratch

    e2n_reduce_kernel<<<B_DIM * M_IN, 256, 0, stream>>>(x, s);
    e2n_wmma_mix_kernel<<<B_DIM * 2 * 12, 32, 0, stream>>>(s, alpha, beta, gamma, out);
}